// ProposalLayer_86517821216530
// MI455X (gfx1250) — compile-verified
//
#include <hip/hip_runtime.h>
#include <hip/hip_bf16.h>
#include <math.h>

// ---------------- problem constants (from reference) ----------------
#define BATCH        8
#define NANCH        262144
#define PRE_NMS      6000
#define PROPOSALS    1000
#define NMS_THR      0.7f

// ---------------- selection machinery ----------------
#define NBUCKET      16384      // top 16 bits of positive-float score
#define CAND_CAP     8192       // >= 6000 + boundary-bucket size (~1K) with slack

// workspace layout (bytes)
#define OFF_HIST     0                                   // B*NBUCKET*4 = 512KB
#define OFF_COUNT    (OFF_HIST + BATCH*NBUCKET*4)        // B*4
#define OFF_THR      (OFF_COUNT + 128)                   // B*4
#define OFF_CSCORE   (OFF_THR + 128)                     // B*CAND_CAP*4
#define OFF_CIDX     (OFF_CSCORE + BATCH*CAND_CAP*4)     // B*CAND_CAP*4
#define OFF_BOXES    (OFF_CIDX + BATCH*CAND_CAP*4)       // B*PRE_NMS*16
#define WS_NEEDED    (OFF_BOXES + BATCH*PRE_NMS*16)

// ---------------- CDNA5 async global->LDS helpers ----------------
// LDS byte address = low 32 bits of the generic pointer (aperture strip,
// per ISA aperture table).
__device__ __forceinline__ unsigned lds_byte_addr(const void* p) {
  return (unsigned)(unsigned long long)p;
}
__device__ __forceinline__ void async_b128_to_lds(unsigned lds_addr, const void* gaddr) {
  unsigned long long ga = (unsigned long long)gaddr;
  asm volatile("global_load_async_to_lds_b128 %0, %1, off"
               :: "v"(lds_addr), "v"(ga) : "memory");
}
__device__ __forceinline__ void wait_async0() {
  asm volatile("s_wait_asynccnt 0" ::: "memory");
}

// ---------------- CDNA5 Tensor Data Mover staging ----------------
#if __has_builtin(__builtin_amdgcn_tensor_load_to_lds)
#define HAVE_TDM 1
typedef unsigned int u32x4 __attribute__((ext_vector_type(4)));
typedef int          i32x4 __attribute__((ext_vector_type(4)));
typedef int          i32x8 __attribute__((ext_vector_type(8)));

// 1-D tile copy of nElem 4-byte words, global -> LDS, via TDM descriptor (D#).
// Group0: count=1 | lds_addr | global_addr[56:0] | type=2 ("image").
// Group1: data_size=4B, tensor_dim0 = tile_dim0 = nElem, multicast mask = 0.
__device__ __forceinline__ void tdm_load_1d(unsigned lds_off, const void* gaddr,
                                            unsigned nElem) {
  unsigned long long ga = (unsigned long long)gaddr;
  u32x4 g0;
  g0.x = 0x1u;                                              // count=1, user mode
  g0.y = lds_off;                                           // lds_addr (bytes)
  g0.z = (unsigned)(ga & 0xFFFFFFFFu);                      // global_addr[31:0]
  g0.w = (unsigned)((ga >> 32) & 0x01FFFFFFu) | (0x2u << 30); // addr[56:32]|type=2
  i32x8 g1;
  g1[0] = 0x00020000;                       // workgroup_mask=0, data_size=2 (4B)
  g1[1] = (int)((nElem & 0xFFFFu) << 16);   // tensor_dim0[15:0]  @ bits 63:48
  g1[2] = (int)((nElem >> 16) & 0xFFFFu);   // tensor_dim0[31:16] @ bits 79:64
  g1[3] = (int)((nElem & 0xFFFFu) << 16);   // tile_dim0          @ bits 127:112
  g1[4] = 0;                                // tile_dim1=0, tile_dim2=0 (1-D)
  g1[5] = (int)nElem;                       // tensor_dim0_stride lo
  g1[6] = 0;
  g1[7] = 0;
  i32x4 gz = {0, 0, 0, 0};
#if defined(__clang_major__) && (__clang_major__ >= 23)
  i32x8 gz8 = {0, 0, 0, 0, 0, 0, 0, 0};
  __builtin_amdgcn_tensor_load_to_lds(g0, g1, gz, gz, gz8, 0);
#else
  __builtin_amdgcn_tensor_load_to_lds(g0, g1, gz, gz, 0);
#endif
}
#endif

// ---------------- K0: zero histogram + counters ----------------
__global__ void k0_zero(unsigned* hist, unsigned* candCount) {
  int i = blockIdx.x * blockDim.x + threadIdx.x;
  if (i < BATCH * NBUCKET) hist[i] = 0u;
  if (i < BATCH) candCount[i] = 0u;
}

// ---------------- K1: histogram of score bit-buckets ----------------
__global__ void k1_hist(const float2* __restrict__ probs, unsigned* __restrict__ hist) {
  int b = blockIdx.y;
  int i = blockIdx.x * blockDim.x + threadIdx.x;
  size_t base = (size_t)b * NANCH;
  if (i + 65536 < NANCH) __builtin_prefetch(&probs[base + i + 65536], 0, 0);
  float score = probs[base + i].y;            // rpn_probs[b, i, 1]
  unsigned bits = __float_as_uint(score);
  unsigned bkt  = bits >> 16;
  if (bkt >= NBUCKET) bkt = NBUCKET - 1;
  atomicAdd(&hist[b * NBUCKET + bkt], 1u);
}

// ---------------- K2: find threshold bucket (suffix count crossing 6000) --------
__global__ __launch_bounds__(1024) void k2_threshold(const unsigned* __restrict__ hist,
                                                     unsigned* __restrict__ thr) {
  int b = blockIdx.x;
  const unsigned* h = hist + (size_t)b * NBUCKET;
  __shared__ unsigned chunk[1024];
  unsigned s = 0;
  int base = threadIdx.x * 16;
  #pragma unroll
  for (int i = 0; i < 16; ++i) s += h[base + i];
  chunk[threadIdx.x] = s;
  __syncthreads();
  if (threadIdx.x == 0) {
    unsigned cum = 0;
    for (int c = 1023; c >= 0; --c) {
      if (cum + chunk[c] >= PRE_NMS) {
        int bb = c * 16 + 15;
        for (; bb >= c * 16; --bb) {
          cum += h[bb];
          if (cum >= PRE_NMS) break;
        }
        thr[b] = (unsigned)(bb < c * 16 ? c * 16 : bb);
        return;
      }
      cum += chunk[c];
    }
    thr[b] = 0u;
  }
}

// ---------------- K3: compact candidates with bucket >= T ----------------
__global__ void k3_compact(const float2* __restrict__ probs,
                           const unsigned* __restrict__ thr,
                           unsigned* __restrict__ candCount,
                           float* __restrict__ candScore,
                           int* __restrict__ candIdx) {
  int b = blockIdx.y;
  int i = blockIdx.x * blockDim.x + threadIdx.x;
  size_t base = (size_t)b * NANCH;
  if (i + 65536 < NANCH) __builtin_prefetch(&probs[base + i + 65536], 0, 0);
  float score = probs[base + i].y;
  unsigned bkt = __float_as_uint(score) >> 16;
  if (bkt >= (unsigned)NBUCKET) bkt = NBUCKET - 1;
  if (bkt >= thr[b]) {
    unsigned pos = atomicAdd(&candCount[b], 1u);
    if (pos < CAND_CAP) {
      candScore[(size_t)b * CAND_CAP + pos] = score;
      candIdx  [(size_t)b * CAND_CAP + pos] = i;
    }
  }
}

// ---------------- K4: bitonic sort 8192 candidates + decode top 6000 boxes -------
// order: score descending, index ascending tiebreak (matches lax.top_k)
__global__ __launch_bounds__(1024) void k4_sort_decode(const float* __restrict__ candScore,
                                                       const int* __restrict__ candIdx,
                                                       const unsigned* __restrict__ candCount,
                                                       const float* __restrict__ rpn_bbox,
                                                       const float* __restrict__ anchors,
                                                       float4* __restrict__ wsBoxes) {
  int b = blockIdx.x;
  extern __shared__ char smem[];
  float* key = (float*)smem;                      // 8192 floats
  int*   val = (int*)(smem + CAND_CAP * 4);       // 8192 ints

  // Async-stage both candidate arrays (64KB) global -> LDS (ASYNCcnt path).
  const float* gs = candScore + (size_t)b * CAND_CAP;
  const int*   gi = candIdx   + (size_t)b * CAND_CAP;
  unsigned keyLds = lds_byte_addr(key);
  unsigned valLds = lds_byte_addr(val);
  for (int c = threadIdx.x; c < CAND_CAP / 4; c += blockDim.x) {
    async_b128_to_lds(keyLds + (unsigned)c * 16u, (const void*)(gs + c * 4));
    async_b128_to_lds(valLds + (unsigned)c * 16u, (const void*)(gi + c * 4));
  }
  wait_async0();
  __syncthreads();

  unsigned cntU = candCount[b];
  int cnt = (cntU > CAND_CAP) ? CAND_CAP : (int)cntU;
  for (int k = cnt + (int)threadIdx.x; k < CAND_CAP; k += blockDim.x) {
    key[k] = -INFINITY;
    val[k] = 0x7FFFFFFF;
  }
  __syncthreads();

  for (int kk = 2; kk <= CAND_CAP; kk <<= 1) {
    for (int j = kk >> 1; j > 0; j >>= 1) {
      for (int t = threadIdx.x; t < CAND_CAP; t += blockDim.x) {
        int ixj = t ^ j;
        if (ixj > t) {
          float ka = key[t], kb = key[ixj];
          int   va = val[t], vb = val[ixj];
          bool prec = (ka > kb) || (ka == kb && va < vb);   // a should precede b
          bool dir  = ((t & kk) == 0);                      // descending segment
          bool sw   = dir ? !prec : prec;
          if (sw) { key[t] = kb; key[ixj] = ka; val[t] = vb; val[ixj] = va; }
        }
      }
      __syncthreads();
    }
  }

  // decode + clip top 6000
  for (int k = threadIdx.x; k < PRE_NMS; k += blockDim.x) {
    int i = val[k];
    float4 box = make_float4(0.f, 0.f, 0.f, 0.f);
    if (i >= 0 && i < NANCH) {
      const float* a = anchors  + ((size_t)b * NANCH + i) * 4;
      const float* d = rpn_bbox + ((size_t)b * NANCH + i) * 4;
      float y1 = a[0], x1 = a[1], y2 = a[2], x2 = a[3];
      float h = y2 - y1, w = x2 - x1;
      float d0 = d[0] * 0.1f, d1 = d[1] * 0.1f, d2 = d[2] * 0.2f, d3 = d[3] * 0.2f;
      float cy = y1 + 0.5f * h + d0 * h;
      float cx = x1 + 0.5f * w + d1 * w;
      h = h * expf(d2);
      w = w * expf(d3);
      float by1 = cy - 0.5f * h, bx1 = cx - 0.5f * w;
      float by2 = cy + 0.5f * h, bx2 = cx + 0.5f * w;
      box.x = fminf(fmaxf(by1, 0.f), 1.f);
      box.y = fminf(fmaxf(bx1, 0.f), 1.f);
      box.z = fminf(fmaxf(by2, 0.f), 1.f);
      box.w = fminf(fmaxf(bx2, 0.f), 1.f);
    }
    wsBoxes[(size_t)b * PRE_NMS + k] = box;
  }
}

// ---------------- K5: exact greedy NMS, windowed (wave32 in-window greedy) -------
// dynamic LDS: boxes 6000*16 = 96000 B, supp flags 6000 B  (~100 KB of 320 KB WGP LDS)
#define NMS_LDS_BYTES (PRE_NMS * 16 + PRE_NMS + 16)
__global__ __launch_bounds__(1024) void k5_nms(const float4* __restrict__ wsBoxes,
                                               float* __restrict__ out) {
  int b = blockIdx.x;
  extern __shared__ char smem[];
  float*         boxes = (float*)smem;                         // [6000][4]
  unsigned char* supp  = (unsigned char*)(smem + PRE_NMS * 16);
  __shared__ int winIdx[32];
  __shared__ int winAcc[32];
  __shared__ int s_m, s_cur, s_kept;

  const float4* src = wsBoxes + (size_t)b * PRE_NMS;

  // ---- stage the box tile global -> LDS ----
#if defined(HAVE_TDM)
  if (threadIdx.x < 32) {           // wave 0 issues one TDM DMA (96000 B)
    tdm_load_1d(lds_byte_addr(boxes), (const void*)src, PRE_NMS * 4u);
    __builtin_amdgcn_s_wait_tensorcnt(0);
  }
#else
  {
    unsigned ldsBase = lds_byte_addr(boxes);
    for (int k = threadIdx.x; k < PRE_NMS; k += blockDim.x)
      async_b128_to_lds(ldsBase + (unsigned)k * 16u, (const void*)(src + k));
    wait_async0();
  }
#endif

  for (int k = threadIdx.x; k < PRE_NMS; k += blockDim.x) supp[k] = 0;
  if (threadIdx.x == 0) { s_cur = 0; s_kept = 0; s_m = 0; }
  __syncthreads();

  float4* out4 = (float4*)(out + (size_t)b * PROPOSALS * 4);

  for (;;) {
    // ---- Phase A: collect next <=32 unsuppressed candidates (in order) ----
    if (threadIdx.x == 0) {
      int c = s_cur, m = 0;
      while (m < 32 && c < PRE_NMS) {
        if (!supp[c]) winIdx[m++] = c;
        ++c;
      }
      s_m = m;
      s_cur = c;
    }
    __syncthreads();
    int m = s_m;
    if (m == 0) break;  // uniform: candidates exhausted

    // ---- Phase B: exact in-window greedy on wave 0 (lane j = candidate j) ----
    if (threadIdx.x < 32) {
      int lane = threadIdx.x;
      bool valid = lane < m;
      int myIdx = valid ? winIdx[lane] : winIdx[0];
      float y1 = boxes[myIdx * 4 + 0], x1 = boxes[myIdx * 4 + 1];
      float y2 = boxes[myIdx * 4 + 2], x2 = boxes[myIdx * 4 + 3];
      float area = (y2 - y1) * (x2 - x1);
      bool alive = valid;
      for (int i = 0; i < m; ++i) {
        int ia = __shfl((int)alive, i);          // was lane i accepted?
        float py1 = __shfl(y1, i), px1 = __shfl(x1, i);
        float py2 = __shfl(y2, i), px2 = __shfl(x2, i);
        float pa  = __shfl(area, i);
        if (ia && lane > i && alive) {
          float yy1 = fmaxf(py1, y1), xx1 = fmaxf(px1, x1);
          float yy2 = fminf(py2, y2), xx2 = fminf(px2, x2);
          float inter = fmaxf(yy2 - yy1, 0.f) * fmaxf(xx2 - xx1, 0.f);
          float uni = pa + area - inter;
          if (inter / fmaxf(uni, 1e-8f) > NMS_THR) alive = false;
        }
      }
      unsigned accMask = (unsigned)__ballot(alive && valid);
      int keptNow = s_kept;
      int rem = PROPOSALS - keptNow;
      int rank = __popc(accMask & ((1u << lane) - 1u));
      bool finalAcc = alive && valid && (rank < rem);  // cap at 1000 picks
      winAcc[lane] = finalAcc ? 1 : 0;
      if (finalAcc) out4[keptNow + rank] = make_float4(y1, x1, y2, x2);
      if (lane == 0) {
        int added = __popc(accMask);
        if (added > rem) added = rem;
        s_kept = keptNow + added;
      }
    }
    __syncthreads();
    if (s_kept >= PROPOSALS) break;  // uniform: reference stops at 1000 picks

    // ---- Phase C: suppress remaining array against this round's accepted ----
    int start = s_cur;
    for (int k = start + (int)threadIdx.x; k < PRE_NMS; k += blockDim.x) {
      if (supp[k]) continue;
      float y1 = boxes[k * 4 + 0], x1 = boxes[k * 4 + 1];
      float y2 = boxes[k * 4 + 2], x2 = boxes[k * 4 + 3];
      float ar = (y2 - y1) * (x2 - x1);
      for (int j = 0; j < m; ++j) {
        if (!winAcc[j]) continue;
        int p = winIdx[j];
        float py1 = boxes[p * 4 + 0], px1 = boxes[p * 4 + 1];
        float py2 = boxes[p * 4 + 2], px2 = boxes[p * 4 + 3];
        float yy1 = fmaxf(py1, y1), xx1 = fmaxf(px1, x1);
        float yy2 = fminf(py2, y2), xx2 = fminf(px2, x2);
        float inter = fmaxf(yy2 - yy1, 0.f) * fmaxf(xx2 - xx1, 0.f);
        float pa = (py2 - py1) * (px2 - px1);
        float uni = pa + ar - inter;
        if (inter / fmaxf(uni, 1e-8f) > NMS_THR) { supp[k] = 1; break; }
      }
    }
    __syncthreads();
  }

  // zero-fill tail rows (reference emits 0.0 for invalid keeps)
  int kept = s_kept;
  for (int idx = kept * 4 + (int)threadIdx.x; idx < PROPOSALS * 4; idx += blockDim.x)
    out[(size_t)b * PROPOSALS * 4 + idx] = 0.f;
}

// ---------------- host entry ----------------
extern "C" void kernel_launch(void* const* d_in, const int* in_sizes, int n_in,
                              void* d_out, int out_size, void* d_ws, size_t ws_size,
                              hipStream_t stream) {
  const float2* rpn_probs = (const float2*)d_in[0];     // (B, N, 2) f32
  const float*  rpn_bbox  = (const float*)d_in[1];      // (B, N, 4) f32
  const float*  anchors   = (const float*)d_in[2];      // (B, N, 4) f32
  float* out = (float*)d_out;                            // (B, 1000, 4) f32
  char*  ws  = (char*)d_ws;

  unsigned* hist      = (unsigned*)(ws + OFF_HIST);
  unsigned* candCount = (unsigned*)(ws + OFF_COUNT);
  unsigned* thr       = (unsigned*)(ws + OFF_THR);
  float*    candScore = (float*)   (ws + OFF_CSCORE);
  int*      candIdx   = (int*)     (ws + OFF_CIDX);
  float4*   wsBoxes   = (float4*)  (ws + OFF_BOXES);

  // allow >64KB dynamic LDS where the runtime wants an opt-in (ignore errors)
  (void)hipFuncSetAttribute((const void*)k4_sort_decode,
                            hipFuncAttributeMaxDynamicSharedMemorySize, CAND_CAP * 8);
  (void)hipFuncSetAttribute((const void*)k5_nms,
                            hipFuncAttributeMaxDynamicSharedMemorySize, NMS_LDS_BYTES);

  dim3 gridStream(NANCH / 256, BATCH);

  k0_zero<<<(BATCH * NBUCKET + 255) / 256, 256, 0, stream>>>(hist, candCount);
  k1_hist<<<gridStream, 256, 0, stream>>>(rpn_probs, hist);
  k2_threshold<<<BATCH, 1024, 0, stream>>>(hist, thr);
  k3_compact<<<gridStream, 256, 0, stream>>>(rpn_probs, thr, candCount, candScore, candIdx);
  k4_sort_decode<<<BATCH, 1024, CAND_CAP * 8, stream>>>(candScore, candIdx, candCount,
                                                        rpn_bbox, anchors, wsBoxes);
  k5_nms<<<BATCH, 1024, NMS_LDS_BYTES, stream>>>(wsBoxes, out);
}